// DCRNNModel_classification_10273561772736
// MI455X (gfx1250) — compile-verified
//
#include <hip/hip_runtime.h>
#include <math.h>

#define N_NODES 19
#define UNITS   64
#define BATCH   256
#define TSTEPS  60
#define D_IN    100

typedef __attribute__((ext_vector_type(2))) float v2f;
typedef __attribute__((ext_vector_type(8))) float v8f;

// ---------------------------------------------------------------------------
// Fused concat + Chebyshev diffusion stack.
// Builds stacked (B*N, 3*Dxh) = [xh | S xh | 2 S(S xh) - xh] column-block order,
// matching the reference's concat-along-last-axis and W's (D*(S*K+1), O) layout.
// One thread handles one (b, d) feature column across all 19 nodes.
// ---------------------------------------------------------------------------
__global__ void diffusion_stack_kernel(
    const float* __restrict__ x, long xBStride, int xNStride,
    const float* __restrict__ h,        // (B, N, 64)
    const float* __restrict__ r,        // nullptr, or gates buffer (row stride 128; r = cols [0,64))
    const float* __restrict__ S,        // (N, N)
    float* __restrict__ stacked,        // (B*N, 3*Dxh)
    int Dx)                             // x feature dim; Dxh = Dx + 64
{
    const int Dxh = Dx + UNITS;
    __shared__ float sS[N_NODES * N_NODES];
    for (int i = threadIdx.x; i < N_NODES * N_NODES; i += blockDim.x)
        sS[i] = S[i];
    __syncthreads();

    const int d = blockIdx.x * blockDim.x + threadIdx.x;
    const int b = blockIdx.y;
    if (d >= Dxh) return;

    float v[N_NODES], y1[N_NODES], y2[N_NODES];
    if (d < Dx) {
        const float* xb = x + (long)b * xBStride;
#pragma unroll
        for (int n = 0; n < N_NODES; ++n) v[n] = xb[(long)n * xNStride + d];
    } else {
        const int dh = d - Dx;
        const float* hb = h + ((long)b * N_NODES) * UNITS + dh;
        if (r) {
            const float* rb = r + ((long)b * N_NODES) * (2 * UNITS) + dh;
#pragma unroll
            for (int n = 0; n < N_NODES; ++n)
                v[n] = hb[n * UNITS] * rb[n * (2 * UNITS)];
        } else {
#pragma unroll
            for (int n = 0; n < N_NODES; ++n) v[n] = hb[n * UNITS];
        }
    }
#pragma unroll
    for (int n = 0; n < N_NODES; ++n) {
        float acc = 0.f;
#pragma unroll
        for (int m = 0; m < N_NODES; ++m) acc += sS[n * N_NODES + m] * v[m];
        y1[n] = acc;
    }
#pragma unroll
    for (int n = 0; n < N_NODES; ++n) {
        float acc = 0.f;
#pragma unroll
        for (int m = 0; m < N_NODES; ++m) acc += sS[n * N_NODES + m] * y1[m];
        y2[n] = 2.f * acc - v[n];
    }
    const long rowStride = 3L * Dxh;
    float* outb = stacked + (long)b * N_NODES * rowStride;
#pragma unroll
    for (int n = 0; n < N_NODES; ++n) {
        float* row = outb + (long)n * rowStride;
        row[d]            = v[n];
        row[Dxh + d]      = y1[n];
        row[2 * Dxh + d]  = y2[n];
    }
}

// ---------------------------------------------------------------------------
// out = act( A(M,K) @ W(K,Nout) + bias ) via V_WMMA_F32_16X16X4_F32.
// One wave per 16x16 output tile; K stepped by 4.
// A fragment (16x4 f32, ISA layout): lane&15 = M row; lane>>4 selects K pair
//   {k0,k0+1} vs {k0+2,k0+3} across the two VGPRs.
// B fragment (4x16 f32): rows striped across lanes, same half-lane K split.
// C/D (16x16 f32, 8 VGPRs): VGPR j -> row j (lanes 0-15) / row 8+j (16-31).
// ---------------------------------------------------------------------------
__global__ void wmma_gemm_act_kernel(
    const float* __restrict__ A, const float* __restrict__ W,
    const float* __restrict__ bias, float* __restrict__ out,
    int M, int K, int Nout, int act /*0=sigmoid, 1=tanh*/)
{
    const int wave   = blockIdx.x * (blockDim.x >> 5) + (threadIdx.x >> 5);
    const int tilesM = M >> 4;
    const int tilesN = Nout >> 4;
    if (wave >= tilesM * tilesN) return;
    const int tm = wave % tilesM;
    const int tn = wave / tilesM;

    const int lane     = threadIdx.x & 31;
    const int lane16   = lane & 15;
    const int laneHalf = lane >> 4;

    const float* arow = A + (long)(tm * 16 + lane16) * K + laneHalf * 2;
    const float* wcol = W + (tn * 16 + lane16) + (long)(laneHalf * 2) * Nout;

    v8f acc = {};
    for (int k0 = 0; k0 < K; k0 += 4) {
        v2f a, bf;
        a.x  = arow[k0];
        a.y  = arow[k0 + 1];
        bf.x = wcol[(long)k0 * Nout];
        bf.y = wcol[(long)(k0 + 1) * Nout];
        acc = __builtin_amdgcn_wmma_f32_16x16x4_f32(
            /*neg_a=*/false, a, /*neg_b=*/false, bf,
            /*c_mod=*/(short)0, acc, /*reuse_a=*/false, /*reuse_b=*/false);
    }

    const int col = tn * 16 + lane16;
    const float bv = bias[col];
#pragma unroll
    for (int j = 0; j < 8; ++j) {
        const int row = tm * 16 + laneHalf * 8 + j;
        float vv = acc[j] + bv;
        vv = (act == 0) ? (1.f / (1.f + __expf(-vv))) : tanhf(vv);
        out[(long)row * Nout + col] = vv;
    }
}

// ---------------------------------------------------------------------------
// GRU combine: h = u*h + (1-u)*c, with optional "last relevant state" capture.
// ---------------------------------------------------------------------------
__global__ void gru_update_kernel(
    float* __restrict__ h, const float* __restrict__ gates,
    const float* __restrict__ c, float* __restrict__ last,
    const int* __restrict__ seq_len, int t, int total)
{
    const int i = blockIdx.x * blockDim.x + threadIdx.x;
    if (i >= total) return;
    const int row = i >> 6;     // /UNITS
    const int j   = i & 63;
    const float u  = gates[(long)row * (2 * UNITS) + UNITS + j];
    const float hn = u * h[i] + (1.f - u) * c[i];
    h[i] = hn;
    if (last) {
        const int b = i / (N_NODES * UNITS);
        int idx = seq_len[b] - 1;
        idx = idx < 0 ? 0 : (idx > TSTEPS - 1 ? TSTEPS - 1 : idx);
        if (idx == t) last[i] = hn;
    }
}

// ---------------------------------------------------------------------------
// logits[b,c] = max_n ( relu(last[b,n,:]) . fc_w[c,:] + fc_b[c] )
// One wave (32 threads) per batch element; lanes 0..18 handle the nodes.
// ---------------------------------------------------------------------------
__global__ void fc_max_kernel(const float* __restrict__ last,
                              const float* __restrict__ fcw,  // (4, 64)
                              const float* __restrict__ fcb,  // (4,)
                              float* __restrict__ out)        // (B, 4)
{
    const int b    = blockIdx.x;
    const int lane = threadIdx.x;   // 0..31
    __shared__ float red[32][4];
    float lg[4] = {-3.4e38f, -3.4e38f, -3.4e38f, -3.4e38f};
    if (lane < N_NODES) {
        const float* hv = last + ((long)b * N_NODES + lane) * UNITS;
        for (int cc = 0; cc < 4; ++cc) {
            float acc = fcb[cc];
            for (int u = 0; u < UNITS; ++u) {
                float x = hv[u];
                x = x > 0.f ? x : 0.f;
                acc += x * fcw[cc * UNITS + u];
            }
            lg[cc] = acc;
        }
    }
    for (int cc = 0; cc < 4; ++cc) red[lane][cc] = lg[cc];
    __syncthreads();
    for (int s = 16; s > 0; s >>= 1) {
        if (lane < s)
            for (int cc = 0; cc < 4; ++cc)
                red[lane][cc] = fmaxf(red[lane][cc], red[lane + s][cc]);
        __syncthreads();
    }
    if (lane < 4) out[b * 4 + lane] = red[0][lane];
}

// ---------------------------------------------------------------------------
extern "C" void kernel_launch(void* const* d_in, const int* in_sizes, int n_in,
                              void* d_out, int out_size, void* d_ws, size_t ws_size,
                              hipStream_t stream) {
    (void)in_sizes; (void)n_in; (void)out_size; (void)ws_size;
    const float* input_seq = (const float*)d_in[0];   // (B, T, N, D)
    const float* supports  = (const float*)d_in[1];   // (1, N, N)
    const float* Wg0 = (const float*)d_in[2];
    const float* bg0 = (const float*)d_in[3];
    const float* Wc0 = (const float*)d_in[4];
    const float* bc0 = (const float*)d_in[5];
    const float* Wg1 = (const float*)d_in[6];
    const float* bg1 = (const float*)d_in[7];
    const float* Wc1 = (const float*)d_in[8];
    const float* bc1 = (const float*)d_in[9];
    const float* fcw = (const float*)d_in[10];
    const float* fcb = (const float*)d_in[11];
    const int*   seqlen = (const int*)d_in[12];

    const int M      = BATCH * N_NODES;        // 4864 rows, /16 exact
    const int hElems = M * UNITS;              // 311296
    const int K0     = 3 * (D_IN + UNITS);     // 492
    const int K1     = 3 * (2 * UNITS);        // 384

    float* ws = (float*)d_ws;
    size_t o = 0;
    float* h0      = ws + o;  o += (size_t)hElems;
    float* h1      = ws + o;  o += (size_t)hElems;
    float* lastbuf = ws + o;  o += (size_t)hElems;
    float* gates   = ws + o;  o += (size_t)M * 2 * UNITS;
    float* cbuf    = ws + o;  o += (size_t)hElems;
    float* stacked = ws + o;  o += (size_t)M * K0;

    // zero both hidden states (h0, h1 contiguous) every call — deterministic
    hipMemsetAsync(h0, 0, 2 * (size_t)hElems * sizeof(float), stream);

    const dim3 dif0Grid((D_IN + UNITS + 63) / 64, BATCH);   // Dxh=164
    const dim3 dif1Grid((2 * UNITS + 63) / 64, BATCH);      // Dxh=128
    const int gemmWaves128 = (M / 16) * (128 / 16);         // 2432
    const int gemmWaves64  = (M / 16) * (64 / 16);          // 1216
    const int gemmBlocks128 = (gemmWaves128 + 3) / 4;
    const int gemmBlocks64  = (gemmWaves64 + 3) / 4;
    const int updBlocks = (hElems + 255) / 256;

    const long  xBStride0 = (long)TSTEPS * N_NODES * D_IN;  // batch stride in input_seq
    const long  xBStride1 = (long)N_NODES * UNITS;          // batch stride in h0

    for (int t = 0; t < TSTEPS; ++t) {
        const float* xt = input_seq + (long)t * N_NODES * D_IN;

        // ---- layer 0 ----
        diffusion_stack_kernel<<<dif0Grid, 64, 0, stream>>>(
            xt, xBStride0, D_IN, h0, nullptr, supports, stacked, D_IN);
        wmma_gemm_act_kernel<<<gemmBlocks128, 128, 0, stream>>>(
            stacked, Wg0, bg0, gates, M, K0, 2 * UNITS, /*sigmoid*/0);
        diffusion_stack_kernel<<<dif0Grid, 64, 0, stream>>>(
            xt, xBStride0, D_IN, h0, gates, supports, stacked, D_IN);
        wmma_gemm_act_kernel<<<gemmBlocks64, 128, 0, stream>>>(
            stacked, Wc0, bc0, cbuf, M, K0, UNITS, /*tanh*/1);
        gru_update_kernel<<<updBlocks, 256, 0, stream>>>(
            h0, gates, cbuf, nullptr, nullptr, t, hElems);

        // ---- layer 1 (input = layer-0 output at this step) ----
        diffusion_stack_kernel<<<dif1Grid, 64, 0, stream>>>(
            h0, xBStride1, UNITS, h1, nullptr, supports, stacked, UNITS);
        wmma_gemm_act_kernel<<<gemmBlocks128, 128, 0, stream>>>(
            stacked, Wg1, bg1, gates, M, K1, 2 * UNITS, /*sigmoid*/0);
        diffusion_stack_kernel<<<dif1Grid, 64, 0, stream>>>(
            h0, xBStride1, UNITS, h1, gates, supports, stacked, UNITS);
        wmma_gemm_act_kernel<<<gemmBlocks64, 128, 0, stream>>>(
            stacked, Wc1, bc1, cbuf, M, K1, UNITS, /*tanh*/1);
        gru_update_kernel<<<updBlocks, 256, 0, stream>>>(
            h1, gates, cbuf, lastbuf, seqlen, t, hElems);
    }

    fc_max_kernel<<<BATCH, 32, 0, stream>>>(lastbuf, fcw, fcb, (float*)d_out);
}